// cluster_MixStyle_51977694216555
// MI455X (gfx1250) — compile-verified
//
#include <hip/hip_runtime.h>
#include <math.h>

typedef __attribute__((ext_vector_type(2))) float v2f;
typedef __attribute__((ext_vector_type(8))) float v8f;

#define BATCH   64
#define CHAN    64
#define HW      12544     // 112*112
#define HW4     3136      // HW/4
#define KCLUST  8
#define EPSF    1e-6f

// ---------------------------------------------------------------------------
// Kernel 1: per-(b,c) plane sum and sum-of-squares. One block per plane,
// float4 (global_load_b128) streaming, LDS tree reduction.
// ---------------------------------------------------------------------------
__global__ void k_plane_stats(const float* __restrict__ x,
                              float* __restrict__ s, float* __restrict__ sq) {
    __shared__ float ls[256];
    __shared__ float lq[256];
    const int plane = blockIdx.x;
    const float4* xv = (const float4*)(x + (size_t)plane * HW);
    float acc_s = 0.0f, acc_q = 0.0f;
    for (int i = threadIdx.x; i < HW4; i += 256) {
        float4 v = xv[i];
        acc_s += v.x + v.y + v.z + v.w;
        acc_q += v.x * v.x + v.y * v.y + v.z * v.z + v.w * v.w;
    }
    ls[threadIdx.x] = acc_s;
    lq[threadIdx.x] = acc_q;
    __syncthreads();
    for (int off = 128; off > 0; off >>= 1) {
        if (threadIdx.x < off) {
            ls[threadIdx.x] += ls[threadIdx.x + off];
            lq[threadIdx.x] += lq[threadIdx.x + off];
        }
        __syncthreads();
    }
    if (threadIdx.x == 0) {
        s[plane]  = ls[0];
        sq[plane] = lq[0];
    }
}

// ---------------------------------------------------------------------------
// Kernel 2 (tiny, 1 block / 8 waves): argmax labels, WMMA-based cluster
// segment sums (D = onehot(16x4) x sums(4x16) accumulated over K=64 batches),
// then fold everything into per-(b,c) scale/shift.
//
// v_wmma_f32_16x16x4_f32 wave32 layouts (ISA 7.12.2):
//   A 16x4 : lane M = lane%16; VGPR0/1 hold K = {0,1} (lanes 0-15) / {2,3} (16-31)
//   B 4x16 : lane N = lane%16; VGPR0/1 hold K = {0,1} (lanes 0-15) / {2,3} (16-31)
//   C/D    : VGPR r -> M = r + 8*(lane>=16), N = lane%16
// ---------------------------------------------------------------------------
__global__ void k_coeffs(const float* __restrict__ cmap,   // [64*8]
                         const float* __restrict__ lmda,   // [64]
                         const float* __restrict__ s,
                         const float* __restrict__ sq,
                         float* __restrict__ Ascale,
                         float* __restrict__ Bshift) {
    __shared__ int   lab[BATCH];
    __shared__ float cnt[KCLUST];
    __shared__ float seg[2 * 16 * CHAN];   // [mat][m=cluster(16)][c]

    const int tid = threadIdx.x;

    // labels: first-occurrence argmax over 8 cluster logits per sample
    if (tid < BATCH) {
        float best = cmap[tid * KCLUST];
        int arg = 0;
        for (int j = 1; j < KCLUST; ++j) {
            float v = cmap[tid * KCLUST + j];
            if (v > best) { best = v; arg = j; }
        }
        lab[tid] = arg;
    }
    __syncthreads();

    // WMMA segment sums. wave -> (mat, ctile): mat 0 = sums, 1 = sumsq.
    const int wave  = tid >> 5;
    const int lane  = tid & 31;
    const int half  = lane >> 4;     // K half selector
    const int lm    = lane & 15;     // A: row m / B: col n
    const int mat   = wave & 1;
    const int ctile = wave >> 1;     // 0..3 -> channels ctile*16..+15
    const float* S  = mat ? sq : s;
    const int c     = ctile * 16 + lm;

    v8f acc = {};
    for (int kb = 0; kb < 16; ++kb) {           // K = batch, 4 per WMMA
        const int b0 = kb * 4 + half * 2;
        v2f av, bv;
        av.x = (lab[b0 + 0] == lm) ? 1.0f : 0.0f;   // onehot(cluster=lm, b)
        av.y = (lab[b0 + 1] == lm) ? 1.0f : 0.0f;
        bv.x = S[(b0 + 0) * CHAN + c];
        bv.y = S[(b0 + 1) * CHAN + c];
        acc = __builtin_amdgcn_wmma_f32_16x16x4_f32(
            /*neg_a=*/false, av, /*neg_b=*/false, bv,
            /*c_mod=*/(short)0, acc, /*reuse_a=*/false, /*reuse_b=*/false);
    }
#pragma unroll
    for (int r = 0; r < 8; ++r) {
        const int m = r + half * 8;             // cluster row (rows 8..15 are 0)
        seg[mat * (16 * CHAN) + m * CHAN + c] = acc[r];
    }

    if (tid < KCLUST) {
        float n = 0.0f;
        for (int b = 0; b < BATCH; ++b) n += (lab[b] == tid) ? 1.0f : 0.0f;
        cnt[tid] = n;
    }
    __syncthreads();

    // fold normalize + mix into out = x*scale + shift per (b,c)
    const float n_s = (float)HW;
    for (int idx = tid; idx < BATCH * CHAN; idx += 256) {
        const int b  = idx >> 6;
        const int ch = idx & 63;
        const float mu   = s[idx] / n_s;
        const float var  = (sq[idx] - n_s * mu * mu) / (n_s - 1.0f);  // ddof=1
        const float sstd = sqrtf(var + EPSF);

        const int   k    = lab[b];
        const float nk   = cnt[k] * n_s;
        const float cmu  = seg[k * CHAN + ch] / nk;
        const float cvar = (seg[16 * CHAN + k * CHAN + ch] - nk * cmu * cmu) / (nk - 1.0f);
        const float cstd = sqrtf(cvar + EPSF);

        const float l       = lmda[b];
        const float mu_mix  = mu * l + cmu * (1.0f - l);
        const float std_mix = sstd * l + cstd * (1.0f - l);
        const float scale   = std_mix / sstd;
        Ascale[idx] = scale;
        Bshift[idx] = mu_mix - mu * scale;
    }
}

// ---------------------------------------------------------------------------
// Kernel 3: out = x*scale[plane] + shift[plane], float4 streaming.
// ---------------------------------------------------------------------------
__global__ void k_apply(const float* __restrict__ x,
                        const float* __restrict__ Ascale,
                        const float* __restrict__ Bshift,
                        float* __restrict__ out) {
    const int plane = blockIdx.x;
    const float a  = Ascale[plane];
    const float bs = Bshift[plane];
    const float4* xv = (const float4*)(x + (size_t)plane * HW);
    float4* ov = (float4*)(out + (size_t)plane * HW);
    for (int i = threadIdx.x; i < HW4; i += 256) {
        float4 v = xv[i];
        float4 r;
        r.x = fmaf(v.x, a, bs);
        r.y = fmaf(v.y, a, bs);
        r.z = fmaf(v.z, a, bs);
        r.w = fmaf(v.w, a, bs);
        ov[i] = r;
    }
}

extern "C" void kernel_launch(void* const* d_in, const int* in_sizes, int n_in,
                              void* d_out, int out_size, void* d_ws, size_t ws_size,
                              hipStream_t stream) {
    const float* x    = (const float*)d_in[0];   // [64,64,112,112]
    const float* cmap = (const float*)d_in[1];   // [1,64,8]
    const float* lmda = (const float*)d_in[2];   // [64]
    float* out = (float*)d_out;

    float* ws = (float*)d_ws;                    // 4 * 4096 floats = 64 KB
    float* s  = ws;
    float* sq = ws + 4096;
    float* As = ws + 8192;
    float* Bs = ws + 12288;

    const int planes = BATCH * CHAN;             // 4096
    k_plane_stats<<<planes, 256, 0, stream>>>(x, s, sq);
    k_coeffs<<<1, 256, 0, stream>>>(cmap, lmda, s, sq, As, Bs);
    k_apply<<<planes, 256, 0, stream>>>(x, As, Bs, out);
}